// Detector_16466904612895
// MI455X (gfx1250) — compile-verified
//
#include <hip/hip_runtime.h>
#include <stdint.h>

// ---------------------------------------------------------------------------
// YOLO decode for MI455X (gfx1250).
// Memory-bound: ~0.99 GB total traffic -> ~42us floor at 23.3 TB/s.
// Strategy: per-workgroup 2D tile DMA (Tensor Data Mover) global->LDS,
// elementwise transform reading LDS in output order (bank-conflict free via
// TDM pad feature: LDS rows of 33 floats), and contiguous ALIGNED float4
// stores (scalar lead-in/tail to fix the odd per-image base offset).
// ---------------------------------------------------------------------------

#define BATCH 64
#define TILE_P 32               // spatial positions per tile
#define NCH    255              // 3 anchors * 85 channels
#define OUT_ROWS 22743          // 76*76*3 + 38*38*3 + 19*19*3
#define OUT_PER_B (OUT_ROWS * 85)

typedef unsigned int v4u __attribute__((ext_vector_type(4)));
typedef int          v8i __attribute__((ext_vector_type(8)));
typedef int          v4i __attribute__((ext_vector_type(4)));

#ifndef __has_builtin
#define __has_builtin(x) 0
#endif

#if __has_builtin(__builtin_amdgcn_tensor_load_to_lds)
#define USE_TDM 1
#else
#define USE_TDM 0
#endif

#if USE_TDM
// Issue a TDM 2D tile load: rows x tile_w f32 elements, global row stride
// row_stride_elems, packed into LDS starting at lds_byte_addr. When pad=1,
// insert 1 DWORD of LDS padding every 32 DWORDs (rows become 33 floats ->
// bank-conflict-free transposed reads). Must be executed by ONE wave only
// via a scalar branch (tensor ops ignore EXEC).
__device__ __forceinline__ void tdm_load_tile(const float* gsrc,
                                              uint32_t lds_byte_addr,
                                              int rows, int tile_w,
                                              int tensor_w_remaining,
                                              long row_stride_elems,
                                              bool pad) {
  const uint64_t ga = (uint64_t)(uintptr_t)gsrc;

  // ---- D# group 0 (128b): count=1, lds_addr, global_addr, type=2 ----
  v4u g0;
  g0.x = 1u;                                     // count=1, is_restore=0
  g0.y = lds_byte_addr;                          // LDS byte address
  g0.z = (uint32_t)ga;                           // global_addr[31:0]
  g0.w = (uint32_t)((ga >> 32) & 0x1FFFFFFull)   // global_addr[56:32]
       | (2u << 30);                             // type = 2 ("image")

  // ---- D# group 1 (256b) ----
  uint32_t w0 = (2u << 16);                      // data_size = 2 -> 4 bytes
  if (pad) w0 |= (1u << 20)                      // pad_enable
              |  (4u << 22)                      // pad_interval: 2^(4+1)=32 DW
              |  (0u << 25);                     // pad_amount: 1 DWORD
  const uint32_t td0 = (uint32_t)tensor_w_remaining;   // tensor_dim0 (OOB)
  const uint32_t td1 = (uint32_t)rows;                 // tensor_dim1
  v8i g1;
  g1[0] = (int)w0;                                     // wg_mask=0 (no cluster)
  g1[1] = (int)((td0 & 0xFFFFu) << 16);                // tensor_dim0[15:0]
  g1[2] = (int)(((td0 >> 16) & 0xFFFFu) | ((td1 & 0xFFFFu) << 16));
  g1[3] = (int)((((uint32_t)tile_w) & 0xFFFFu) << 16); // tile_dim0
  g1[4] = (int)((uint32_t)rows & 0xFFFFu);             // tile_dim1; tile_dim2=0
  g1[5] = (int)(uint32_t)(row_stride_elems & 0xFFFFFFFFl); // dim0_stride lo32
  g1[6] = (int)(uint32_t)((row_stride_elems >> 32) & 0xFFFFl); // hi16; dim1_stride=0
  g1[7] = 0;

  // ---- groups 2/3: 2D tile; unused higher tensor dims set to 1 so any OOB
  //      check on them passes trivially ----
  v4i g2 = {1, 1, 0, 0};          // tensor_dim2=1, tensor_dim3=1, tile_dim3=0
  v4i g3 = {0, 1 << 16, 0, 0};    // tensor_dim4=1, tile_dim4=0

#if defined(__clang_major__) && (__clang_major__ >= 23)
  v8i gz8 = {0, 0, 0, 0, 0, 0, 0, 0};
  __builtin_amdgcn_tensor_load_to_lds(g0, g1, g2, g3, gz8, 0);
#else
  __builtin_amdgcn_tensor_load_to_lds(g0, g1, g2, g3, 0);
#endif
}
#endif // USE_TDM

// Decode one output element j of a tile (j = p*255 + c, c = a*85 + ch).
template <int S>
__device__ __forceinline__ float decode_one(const float* __restrict__ buf,
                                            int j, int rowstride, int pos0,
                                            float aw0, float ah0, float aw1,
                                            float ah1, float aw2, float ah2) {
  constexpr float STRIDE = (float)(608 / S);
  const int p   = j / NCH;               // constant-divisor magic
  const int c   = j - p * NCH;
  const int a   = c / 85;
  const int ch  = c - a * 85;
  const int pos = pos0 + p;

  const float v  = buf[c * rowstride + p];
  const float e  = __expf(-v);                        // exp(-v)
  const float sg = __builtin_amdgcn_rcpf(1.0f + e);   // sigmoid(v)

  float r;
  if (ch >= 4) {
    r = sg;                                           // conf/cls
  } else if (ch < 2) {
    const int g = (ch == 0) ? (pos % S) : (pos / S);  // gx / gy (S constexpr)
    r = (sg * 1.05f - 0.025f + (float)g) * STRIDE;    // xy
  } else {
    float an;
    if (ch == 2) an = (a == 0) ? aw0 : ((a == 1) ? aw1 : aw2);
    else         an = (a == 0) ? ah0 : ((a == 1) ? ah1 : ah2);
    r = an * __builtin_amdgcn_rcpf(e);                // exp(v)*anchor (wh)
  }
  return r;
}

// One workgroup: TILE_P spatial positions x 255 channels of one image.
// Output block out[b, ROW_OFF + pos0*3 .. , :] is a single contiguous run of
// rem*255 floats (since (pos*3+a)*85+ch == pos*255 + (a*85+ch)).
template <int S, int ROW_OFF>
__global__ __launch_bounds__(256, 1) void yolo_decode_tdm(
    const float* __restrict__ in, float* __restrict__ out,
    float aw0, float ah0, float aw1, float ah1, float aw2, float ah2) {
  constexpr int S2 = S * S;
  constexpr int NT = (S2 + TILE_P - 1) / TILE_P;

  __shared__ float buf[NCH * (TILE_P + 1)];   // 255*33 floats = 33,660 B

  const int tile = (int)blockIdx.x % NT;
  const int b    = (int)blockIdx.x / NT;
  const int pos0 = tile * TILE_P;
  const int left = S2 - pos0;
  const int rem  = (left < TILE_P) ? left : TILE_P;
  const bool full = (rem == TILE_P);

  const float* src = in + (size_t)b * NCH * S2 + pos0;

#if USE_TDM
  // Tensor ops ignore EXEC -> skip via a *scalar* branch so exactly one wave
  // issues the DMA.
  const int wave = __builtin_amdgcn_readfirstlane((int)(threadIdx.x >> 5));
  if (wave == 0) {
    tdm_load_tile(src, (uint32_t)(uintptr_t)&buf[0],
                  NCH, rem, left, (long)S2, full);
    __builtin_amdgcn_s_wait_tensorcnt(0);
  }
#else
  // Fallback: cooperative coalesced load (c-major, p fastest).
  {
    const int rowstride = full ? (TILE_P + 1) : rem;
    for (int i = (int)threadIdx.x; i < rem * NCH; i += 256) {
      const int c = i / rem;
      const int p = i - c * rem;
      buf[c * rowstride + p] = src[(size_t)c * S2 + p];
    }
  }
#endif
  __syncthreads();

  const size_t base = (size_t)b * OUT_PER_B + (size_t)(ROW_OFF + pos0 * 3) * 85;
  const int    t    = (int)threadIdx.x;

  if (full) {
    // ---- main path: compile-time N, LDS row stride 33, float4 stores ----
    constexpr int N  = TILE_P * NCH;              // 8160
    constexpr int RS = TILE_P + 1;                // 33
    // Per-image base is an odd float offset; align the vector stores.
    const int shift = (int)((4 - (base & 3)) & 3);      // 0..3 lead elements
    const int nv    = (N - shift) >> 2;                 // aligned float4 count
    const int tailn = (N - shift) & 3;                  // 0..3 tail elements
    const int tail0 = shift + nv * 4;

    if (t < shift)
      out[base + t] = decode_one<S>(buf, t, RS, pos0,
                                    aw0, ah0, aw1, ah1, aw2, ah2);
    if (t >= 4 && t < 4 + tailn) {
      const int j = tail0 + (t - 4);
      out[base + j] = decode_one<S>(buf, j, RS, pos0,
                                    aw0, ah0, aw1, ah1, aw2, ah2);
    }
    for (int vq = t; vq < nv; vq += 256) {
      const int j0 = shift + vq * 4;
      float4 o;
      o.x = decode_one<S>(buf, j0 + 0, RS, pos0, aw0, ah0, aw1, ah1, aw2, ah2);
      o.y = decode_one<S>(buf, j0 + 1, RS, pos0, aw0, ah0, aw1, ah1, aw2, ah2);
      o.z = decode_one<S>(buf, j0 + 2, RS, pos0, aw0, ah0, aw1, ah1, aw2, ah2);
      o.w = decode_one<S>(buf, j0 + 3, RS, pos0, aw0, ah0, aw1, ah1, aw2, ah2);
      *reinterpret_cast<float4*>(out + base + j0) = o;  // global_store_b128
    }
  } else {
    // ---- rare edge tiles (rem < 32): scalar stores, packed LDS rows ----
    const int n = rem * NCH;
    for (int j = t; j < n; j += 256)
      out[base + j] = decode_one<S>(buf, j, rem, pos0,
                                    aw0, ah0, aw1, ah1, aw2, ah2);
  }
}

extern "C" void kernel_launch(void* const* d_in, const int* in_sizes, int n_in,
                              void* d_out, int out_size, void* d_ws, size_t ws_size,
                              hipStream_t stream) {
  (void)in_sizes; (void)n_in; (void)out_size; (void)d_ws; (void)ws_size;
  const float* x0 = (const float*)d_in[0];   // [64, 255, 76, 76]
  const float* x1 = (const float*)d_in[1];   // [64, 255, 38, 38]
  const float* x2 = (const float*)d_in[2];   // [64, 255, 19, 19]
  float* out = (float*)d_out;                // [64, 22743, 85]

  constexpr int NT0 = (76 * 76 + TILE_P - 1) / TILE_P;  // 181
  constexpr int NT1 = (38 * 38 + TILE_P - 1) / TILE_P;  // 46
  constexpr int NT2 = (19 * 19 + TILE_P - 1) / TILE_P;  // 12

  yolo_decode_tdm<76, 0><<<BATCH * NT0, 256, 0, stream>>>(
      x0, out, 28.f, 28.f, 46.f, 45.f, 64.f, 66.f);
  yolo_decode_tdm<38, 17328><<<BATCH * NT1, 256, 0, stream>>>(
      x1, out, 102.f, 74.f, 78.f, 115.f, 132.f, 113.f);
  yolo_decode_tdm<19, 21660><<<BATCH * NT2, 256, 0, stream>>>(
      x2, out, 149.f, 163.f, 174.f, 268.f, 257.f, 176.f);
}